// Interaction_GraphConvolution_55963423867449
// MI455X (gfx1250) — compile-verified
//
#include <hip/hip_runtime.h>

typedef __attribute__((ext_vector_type(2))) float v2f;
typedef __attribute__((ext_vector_type(8))) float v8f;

#define NNODES 100000
#define NEDGES 1600000
#define FDIM   64

// ---------------------------------------------------------------------------
// Utility: zero a buffer (float4 granularity)
// ---------------------------------------------------------------------------
__global__ __launch_bounds__(256) void k_zero4(float4* __restrict__ p, long n4) {
  long i = (long)blockIdx.x * blockDim.x + threadIdx.x;
  if (i < n4) p[i] = make_float4(0.f, 0.f, 0.f, 0.f);
}

// ---------------------------------------------------------------------------
// deg_brother pipeline: db = segsum(vals_u * degree[col]) - degree ; 0 -> 1
// stored as reciprocal for the fused SpMM scale.
// ---------------------------------------------------------------------------
__global__ __launch_bounds__(256) void k_init_db(const float* __restrict__ degree,
                                                 float* __restrict__ db, int n) {
  int i = blockIdx.x * blockDim.x + threadIdx.x;
  if (i < n) db[i] = -degree[i];
}

__global__ __launch_bounds__(256) void k_edge_db(const int* __restrict__ rows,
                                                 const int* __restrict__ cols,
                                                 const float* __restrict__ evals,
                                                 const float* __restrict__ degree,
                                                 float* __restrict__ db, int e) {
  int i = blockIdx.x * blockDim.x + threadIdx.x;
  if (i >= e) return;
  int r = rows[i];
  float vu = degree[r] * evals[i];
  atomicAdd(&db[r], vu * degree[cols[i]]);
}

__global__ __launch_bounds__(256) void k_fin_db(float* __restrict__ db, int n) {
  int i = blockIdx.x * blockDim.x + threadIdx.x;
  if (i >= n) return;
  float d = db[i];
  db[i] = (d == 0.f) ? 1.f : (1.f / d);   // inv(deg_brother), 0 -> 1
}

// ---------------------------------------------------------------------------
// Dense GEMM: Wf = X @ W + b  via V_WMMA_F32_16X16X4_F32.
// One wave computes a 16x64 output strip; 8 waves (8 row-tiles) per block.
// W (64x64 f32, 16 KB) + b staged in LDS.
// ---------------------------------------------------------------------------
__global__ __launch_bounds__(256) void k_gemm_wf(const float* __restrict__ X,
                                                 const float* __restrict__ W,
                                                 const float* __restrict__ bias,
                                                 float* __restrict__ Wf, int nrows) {
  __shared__ float sW[FDIM * FDIM];
  __shared__ float sB[FDIM];

  const int tid = threadIdx.x;
  const float4* W4 = (const float4*)W;
  float4* sW4 = (float4*)sW;
  for (int i = tid; i < FDIM * FDIM / 4; i += 256) sW4[i] = W4[i];
  if (tid < FDIM) sB[tid] = bias[tid];
  __syncthreads();

  const int wave = tid >> 5;
  const int lane = tid & 31;
  const int half = lane >> 4;   // 0: lanes 0-15, 1: lanes 16-31
  const int l16  = lane & 15;

  const int rowTile = blockIdx.x * 8 + wave;
  const int base = rowTile * 16;
  if (base >= nrows) return;

  const float* Xrow = X + (size_t)(base + l16) * FDIM;

  v8f acc0 = {}, acc1 = {}, acc2 = {}, acc3 = {};

  #pragma unroll
  for (int k = 0; k < FDIM; k += 4) {
    const int ka = k + half * 2;      // A layout: lanes 0-15 -> K=k,k+1 ; 16-31 -> K=k+2,k+3
    v2f a;
    a.x = Xrow[ka];
    a.y = Xrow[ka + 1];

    v2f b0, b1, b2, b3;               // B layout mirrors A: row K striped by lane-half
    b0.x = sW[ka * FDIM +  0 + l16];  b0.y = sW[(ka + 1) * FDIM +  0 + l16];
    b1.x = sW[ka * FDIM + 16 + l16];  b1.y = sW[(ka + 1) * FDIM + 16 + l16];
    b2.x = sW[ka * FDIM + 32 + l16];  b2.y = sW[(ka + 1) * FDIM + 32 + l16];
    b3.x = sW[ka * FDIM + 48 + l16];  b3.y = sW[(ka + 1) * FDIM + 48 + l16];

    acc0 = __builtin_amdgcn_wmma_f32_16x16x4_f32(false, a, false, b0, (short)0, acc0, false, false);
    acc1 = __builtin_amdgcn_wmma_f32_16x16x4_f32(false, a, false, b1, (short)0, acc1, false, false);
    acc2 = __builtin_amdgcn_wmma_f32_16x16x4_f32(false, a, false, b2, (short)0, acc2, false, false);
    acc3 = __builtin_amdgcn_wmma_f32_16x16x4_f32(false, a, false, b3, (short)0, acc3, false, false);
  }

  // C/D layout: VGPR v holds row (base + v + half*8), column tile*16 + l16.
  v8f accs[4] = {acc0, acc1, acc2, acc3};
  #pragma unroll
  for (int t = 0; t < 4; ++t) {
    const int col = t * 16 + l16;
    const float bv = sB[col];
    #pragma unroll
    for (int v = 0; v < 8; ++v) {
      const int row = base + half * 8 + v;
      Wf[(size_t)row * FDIM + col] = accs[t][v] + bv;
    }
  }
}

// ---------------------------------------------------------------------------
// SpMM with fused per-row scaling: Y[r] += (evals[e] * s0[r]? * s1[r]?) * X[c]
// One wave per edge: 32 lanes x float2 = 64 columns. Atomic scatter-add.
// ---------------------------------------------------------------------------
__global__ __launch_bounds__(256) void k_spmm(const int* __restrict__ rows,
                                              const int* __restrict__ cols,
                                              const float* __restrict__ evals,
                                              const float* __restrict__ s0,
                                              const float* __restrict__ s1,
                                              const float* __restrict__ X,
                                              float* __restrict__ Y, int nE) {
  const long gid = (long)blockIdx.x * blockDim.x + threadIdx.x;
  const long e = gid >> 5;
  if (e >= nE) return;
  const int lane = threadIdx.x & 31;

  const int r = rows[e];
  const int c = cols[e];
  float v = evals[e];
  if (s0) v *= s0[r];
  if (s1) v *= s1[r];

  const float2 x = ((const float2*)(X + (size_t)c * FDIM))[lane];
  float* yp = Y + (size_t)r * FDIM + lane * 2;
  atomicAdd(yp,     v * x.x);
  atomicAdd(yp + 1, v * x.y);
}

// ---------------------------------------------------------------------------
// M = Wf * (temp2 / deg_safe - Wf)   (float4 granularity)
// ---------------------------------------------------------------------------
__global__ __launch_bounds__(256) void k_make_m(const float* __restrict__ Wf,
                                                const float* __restrict__ t2,
                                                const float* __restrict__ degree,
                                                float* __restrict__ M, long n4) {
  long i = (long)blockIdx.x * blockDim.x + threadIdx.x;
  if (i >= n4) return;
  const long row = i >> 4;   // 16 float4 per row of 64
  float d = degree[row];
  const float inv = (d == 0.f) ? 1.f : (1.f / d);
  const float4 wf = ((const float4*)Wf)[i];
  const float4 t  = ((const float4*)t2)[i];
  float4 m;
  m.x = wf.x * (t.x * inv - wf.x);
  m.y = wf.y * (t.y * inv - wf.y);
  m.z = wf.z * (t.z * inv - wf.z);
  m.w = wf.w * (t.w * inv - wf.w);
  ((float4*)M)[i] = m;
}

// ---------------------------------------------------------------------------
extern "C" void kernel_launch(void* const* d_in, const int* in_sizes, int n_in,
                              void* d_out, int out_size, void* d_ws, size_t ws_size,
                              hipStream_t stream) {
  const float* X      = (const float*)d_in[0];  // [N,64]
  const int*   rows   = (const int*)  d_in[1];  // [E]
  const int*   cols   = (const int*)  d_in[2];  // [E]
  const float* evals  = (const float*)d_in[3];  // [E]
  const float* degree = (const float*)d_in[4];  // [N]
  const float* W      = (const float*)d_in[5];  // [64,64]
  const float* bvec   = (const float*)d_in[6];  // [64]
  float* out = (float*)d_out;                   // [N,64]

  // Workspace carve-out
  char* ws = (char*)d_ws;
  size_t off = 0;
  auto carve = [&](size_t bytes) -> void* {
    void* p = ws + off;
    off += (bytes + 255) & ~(size_t)255;
    return p;
  };
  const size_t featBytes = (size_t)NNODES * FDIM * sizeof(float);
  float* db = (float*)carve((size_t)NNODES * sizeof(float)); // deg_brother -> inv
  float* Wf = (float*)carve(featBytes);
  float* t1 = (float*)carve(featBytes);   // temp1; later reused as M buffer
  float* t2 = (float*)carve(featBytes);

  const long n4feat = (long)NNODES * (FDIM / 4);     // float4 count of a feature buf
  const int  nodeBlocks = (NNODES + 255) / 256;
  const int  edgeBlocks = (NEDGES + 255) / 256;
  const int  featBlocks = (int)((n4feat + 255) / 256);
  const long spmmThreads = (long)NEDGES * 32;
  const int  spmmBlocks = (int)((spmmThreads + 255) / 256);
  const int  gemmBlocks = (NNODES / 16 + 7) / 8;     // 6250 row-tiles, 8 waves/block

  // Zero accumulation targets (harness poisons them)
  k_zero4<<<featBlocks, 256, 0, stream>>>((float4*)t1,  n4feat);
  k_zero4<<<featBlocks, 256, 0, stream>>>((float4*)t2,  n4feat);
  k_zero4<<<featBlocks, 256, 0, stream>>>((float4*)out, n4feat);

  // deg_brother -> inverse
  k_init_db<<<nodeBlocks, 256, 0, stream>>>(degree, db, NNODES);
  k_edge_db<<<edgeBlocks, 256, 0, stream>>>(rows, cols, evals, degree, db, NEDGES);
  k_fin_db <<<nodeBlocks, 256, 0, stream>>>(db, NNODES);

  // Wf = X @ W + b  (WMMA f32 16x16x4)
  k_gemm_wf<<<gemmBlocks, 256, 0, stream>>>(X, W, bvec, Wf, NNODES);

  // temp1 = spmm(vals_u, Wf)          vals_u = degree[r] * evals
  k_spmm<<<spmmBlocks, 256, 0, stream>>>(rows, cols, evals, degree, nullptr, Wf, t1, NEDGES);
  // temp2 = spmm(vals_b, temp1)       vals_b = degree[r] * evals * inv_db[r]
  k_spmm<<<spmmBlocks, 256, 0, stream>>>(rows, cols, evals, degree, db, t1, t2, NEDGES);

  // M (reuse t1) = Wf * (temp2/deg_safe - Wf)
  k_make_m<<<featBlocks, 256, 0, stream>>>(Wf, t2, degree, t1, n4feat);

  // out = spmm(evals, M)
  k_spmm<<<spmmBlocks, 256, 0, stream>>>(rows, cols, evals, nullptr, nullptr, t1, out, NEDGES);
}